// CascadeLayer_10153302687980
// MI455X (gfx1250) — compile-verified
//
#include <hip/hip_runtime.h>

// Problem constants (match reference)
#define NN 50000
#define NE 800000
#define CC 128
#define RT 5            // row-tiles (16 rows each) per workgroup: 80 rows; 50000 = 625*80
#define PITCH 132       // LDS row pitch (floats): banks 4m mod 64 distinct, 8B/16B aligned

typedef __attribute__((ext_vector_type(2))) float v2f;
typedef __attribute__((ext_vector_type(8))) float v8f;

// ---------------- elementwise helpers ----------------
__global__ void k_zero4(float* __restrict__ p, int n4) {
  int i = blockIdx.x * blockDim.x + threadIdx.x;
  if (i < n4) ((float4*)p)[i] = make_float4(0.f, 0.f, 0.f, 0.f);
}

__global__ void k_scalecopy4(const float* __restrict__ in, float* __restrict__ out,
                             float a, int n4) {
  int i = blockIdx.x * blockDim.x + threadIdx.x;
  if (i < n4) {
    float4 v = ((const float4*)in)[i];
    ((float4*)out)[i] = make_float4(a * v.x, a * v.y, a * v.z, a * v.w);
  }
}

// ---------------- degree + symmetric normalization ----------------
__global__ void k_deg(const int* __restrict__ row, const float* __restrict__ w,
                      float* __restrict__ deg) {
  int e = blockIdx.x * blockDim.x + threadIdx.x;
  if (e < NE) atomicAdd(&deg[row[e]], w[e]);
}

__global__ void k_norm(const int* __restrict__ row, const int* __restrict__ col,
                       const float* __restrict__ w, const float* __restrict__ deg,
                       float* __restrict__ norm) {
  int e = blockIdx.x * blockDim.x + threadIdx.x;
  if (e >= NE) return;
  float dr = deg[row[e]], dc = deg[col[e]];
  float ir = dr > 0.f ? rsqrtf(dr) : 0.f;
  float ic = dc > 0.f ? rsqrtf(dc) : 0.f;
  norm[e] = ir * w[e] * ic;
}

// ---------------- scaled SpMM scatter: hout[col] += scale*norm*hin[row] --------
// One wave32 per edge; lane handles 4 channels (float4 gather, 4 f32 atomics).
__global__ void k_spmm(const float* __restrict__ hin, float* __restrict__ hout,
                       const int* __restrict__ row, const int* __restrict__ col,
                       const float* __restrict__ norm, float scale) {
  int e = blockIdx.x * (blockDim.x >> 5) + (threadIdx.x >> 5);
  if (e >= NE) return;
  int lane = threadIdx.x & 31;
  int r = row[e], c = col[e];
  float s = scale * norm[e];
  float4 v = ((const float4*)(hin + (size_t)r * CC))[lane];
  float* dst = hout + (size_t)c * CC + lane * 4;
  atomicAdd(dst + 0, s * v.x);
  atomicAdd(dst + 1, s * v.y);
  atomicAdd(dst + 2, s * v.z);
  atomicAdd(dst + 3, s * v.w);
}

// ---------------- weight pre-pack: Wt[s][cout][cin] = W_{head(s)}[k(s)][cin][cout] ---
// Slice order s = head*(head+1)/2 + k  (10 slices). Makes each B fragment one b64 load.
__global__ void k_packW(const float* __restrict__ W1, const float* __restrict__ W2,
                        const float* __restrict__ W3, const float* __restrict__ W4,
                        float* __restrict__ Wt) {
  int id = blockIdx.x * blockDim.x + threadIdx.x;
  if (id >= 10 * CC * CC) return;
  int s = id >> 14;          // / (128*128)
  int rem = id & 16383;
  int c = rem >> 7;          // CIN index (contiguous reads over d)
  int d = rem & 127;         // COUT index
  int h = (s >= 6) ? 3 : (s >= 3) ? 2 : (s >= 1) ? 1 : 0;
  int k = s - (h * (h + 1)) / 2;
  const float* W = (h == 0) ? W1 : (h == 1) ? W2 : (h == 2) ? W3 : W4;
  Wt[((size_t)s * CC + d) * CC + c] = W[((size_t)k * CC + c) * CC + d];
}

// ---------------- fused 4-head GEMM with V_WMMA_F32_16X16X4_F32 ----------------
// Grid: 625 WGs x 256 threads. Wave = column tile (8 ntiles). Each WG covers 80 rows
// (5 row-tiles) so every B fragment is reused 5x. A tiles for all 4 bases live in
// 169KB dynamic LDS (pitch 132 -> conflict-free ds_load_b64).
// Fragment layouts per CDNA5 ISA 7.12.2:
//   A (16x4 f32): lane 0-15 M=lane {v0:K, v1:K+1}; lanes 16-31 K+=2
//   B (4x16 f32): lane -> N, same K split across lane halves
//   C/D: VGPR j -> M = j + 8*(lane>>4), N = lane&15
__global__ __launch_bounds__(256) void k_gemm(
    const float* __restrict__ T0, const float* __restrict__ T1,
    const float* __restrict__ T2, const float* __restrict__ T3,
    const float* __restrict__ Wt,
    const float* __restrict__ b1, const float* __restrict__ b2,
    const float* __restrict__ b3, const float* __restrict__ b4,
    float* __restrict__ out) {
  extern __shared__ float sA[];  // [4][RT*16][PITCH]
  const int t = threadIdx.x;
  const int rowbase = blockIdx.x * (RT * 16);

  // Stage 4 x 80 x 128 fp32 = 10240 float4s; 40 per thread.
  for (int it = 0; it < 40; ++it) {
    int q = it * 256 + t;
    int kb = q / 2560;               // basis (80*128/4 = 2560 float4 each)
    int rem = q - kb * 2560;
    int r = rem >> 5;                // row 0..79
    int c4 = rem & 31;               // float4 column
    const float* base = (kb == 0) ? T0 : (kb == 1) ? T1 : (kb == 2) ? T2 : T3;
    float4 v = ((const float4*)(base + (size_t)(rowbase + r) * CC))[c4];
    float* d = &sA[(kb * (RT * 16) + r) * PITCH + c4 * 4];
    d[0] = v.x; d[1] = v.y; d[2] = v.z; d[3] = v.w;
  }
  __syncthreads();

  const int wv = t >> 5;             // wave id == column tile
  const int lane = t & 31;
  const int kh = lane >> 4;
  const int nl = lane & 15;
  const int n0 = wv * 16;

  v8f acc[RT][4] = {};               // 5 row-tiles x 4 heads (160 VGPRs)

  for (int kk = 0; kk < CC; kk += 4) {
    const int kb = kk + 2 * kh;
    // Load all 10 B fragments once per kk (single b64 each from packed Wt).
    v2f B[10];
#pragma unroll
    for (int s = 0; s < 10; ++s) {
      float2 bv = *(const float2*)&Wt[((size_t)s * CC + n0 + nl) * CC + kb];
      B[s].x = bv.x; B[s].y = bv.y;
    }
#pragma unroll
    for (int r = 0; r < RT; ++r) {
#pragma unroll
      for (int k = 0; k < 4; ++k) {
        float2 av = *(const float2*)&sA[((k * RT + r) * 16 + nl) * PITCH + kb];
        v2f a; a.x = av.x; a.y = av.y;
#pragma unroll
        for (int i = k; i < 4; ++i) {     // head i uses basis k iff k <= i
          const int s = (i * (i + 1)) / 2 + k;
          acc[r][i] = __builtin_amdgcn_wmma_f32_16x16x4_f32(
              false, a, false, B[s], (short)0, acc[r][i], false, false);
        }
      }
    }
  }

  // bias + relu + store
  const float* bp[4] = {b1, b2, b3, b4};
#pragma unroll
  for (int i = 0; i < 4; ++i) {
    float bias = bp[i][n0 + nl];
#pragma unroll
    for (int r = 0; r < RT; ++r) {
#pragma unroll
      for (int j = 0; j < 8; ++j) {
        int m = r * 16 + j + 8 * kh;
        float v = acc[r][i][j] + bias;
        out[(size_t)i * NN * CC + (size_t)(rowbase + m) * CC + n0 + nl] =
            v > 0.f ? v : 0.f;
      }
    }
  }
}

// ---------------- orchestration ----------------
extern "C" void kernel_launch(void* const* d_in, const int* in_sizes, int n_in,
                              void* d_out, int out_size, void* d_ws, size_t ws_size,
                              hipStream_t stream) {
  const float* x  = (const float*)d_in[0];
  const int* eidx = (const int*)d_in[1];
  const int* row  = eidx;        // edge_index[0]
  const int* col  = eidx + NE;   // edge_index[1]
  const float* w  = (const float*)d_in[2];
  const float* W1 = (const float*)d_in[3];
  const float* b1 = (const float*)d_in[4];
  const float* W2 = (const float*)d_in[5];
  const float* b2 = (const float*)d_in[6];
  const float* W3 = (const float*)d_in[7];
  const float* b3 = (const float*)d_in[8];
  const float* W4 = (const float*)d_in[9];
  const float* b4 = (const float*)d_in[10];
  float* out = (float*)d_out;

  // workspace carve-up (256B aligned)
  char* ws = (char*)d_ws;
  auto carve = [&ws](size_t bytes) {
    void* p = (void*)ws;
    ws += (bytes + 255) & ~(size_t)255;
    return p;
  };
  float* deg  = (float*)carve((size_t)NN * 4);
  float* norm = (float*)carve((size_t)NE * 4);
  float* T1   = (float*)carve((size_t)NN * CC * 4);
  float* T2   = (float*)carve((size_t)NN * CC * 4);
  float* T3   = (float*)carve((size_t)NN * CC * 4);
  float* Wt   = (float*)carve((size_t)10 * CC * CC * 4);

  const int n4feat = NN * CC / 4;
  const int n4deg  = NN / 4;

  // weight pre-pack (independent of graph pipeline)
  k_packW<<<(10 * CC * CC + 255) / 256, 256, 0, stream>>>(W1, W2, W3, W4, Wt);

  // deg / norm
  k_zero4<<<(n4deg + 255) / 256, 256, 0, stream>>>(deg, n4deg);
  k_deg<<<(NE + 255) / 256, 256, 0, stream>>>(row, w, deg);
  k_norm<<<(NE + 255) / 256, 256, 0, stream>>>(row, col, w, deg, norm);

  // T1 = -prop(x)              (= 2*alpha*lhat(x), alpha=0.5, lhat = -prop)
  k_zero4<<<(n4feat + 255) / 256, 256, 0, stream>>>(T1, n4feat);
  k_spmm<<<NE / 8, 256, 0, stream>>>(x, T1, row, col, norm, -1.0f);

  // T2 = -1.5*prop(T1) - 0.5*T0
  k_scalecopy4<<<(n4feat + 255) / 256, 256, 0, stream>>>(x, T2, -0.5f, n4feat);
  k_spmm<<<NE / 8, 256, 0, stream>>>(T1, T2, row, col, norm, -1.5f);

  // T3 = -(5/3)*prop(T2) - (2/3)*T1
  k_scalecopy4<<<(n4feat + 255) / 256, 256, 0, stream>>>(T1, T3, -2.0f / 3.0f, n4feat);
  k_spmm<<<NE / 8, 256, 0, stream>>>(T2, T3, row, col, norm, -5.0f / 3.0f);

  // fused 4-head WMMA GEMM: 625 WGs x 80 rows, 169KB dynamic LDS
  const size_t ldsBytes = (size_t)4 * (RT * 16) * PITCH * sizeof(float);
  k_gemm<<<NN / (RT * 16), 256, ldsBytes, stream>>>(x, T1, T2, T3, Wt,
                                                    b1, b2, b3, b4, out);
}